// SpAccAvModule_9732395893062
// MI455X (gfx1250) — compile-verified
//
#include <hip/hip_runtime.h>

typedef float v2f __attribute__((ext_vector_type(2)));
typedef float v8f __attribute__((ext_vector_type(8)));

#define CHANNELS 32

// ---------------------------------------------------------------------------
// Native CDNA5 f32 atomic add, no-return form (tracked by STOREcnt only),
// device scope so the RMW happens at the L2 point of coherence.
// ---------------------------------------------------------------------------
__device__ __forceinline__ void atomic_add_f32_dev(float* p, float v) {
  asm volatile("global_atomic_add_f32 %0, %1, off scope:SCOPE_DEV"
               :
               : "v"(p), "v"(v)
               : "memory");
}

// ---------------------------------------------------------------------------
// Zero a float buffer (graph-capture safe; runs every launch for determinism).
// ---------------------------------------------------------------------------
__global__ void zero_f32(float* __restrict__ p, int n) {
  int i = blockIdx.x * blockDim.x + threadIdx.x;
  int stride = gridDim.x * blockDim.x;
  for (; i < n; i += stride) p[i] = 0.0f;
}

// ---------------------------------------------------------------------------
// Finalize phase: avg = diag(1/max(count,1)) * sums, done as WMMA f32 16x16x4.
// Each wave owns a 16-segment x 32-channel tile (two 16x16 column tiles).
// A-slab layout (32-bit A 16x4): lanes 0-15 -> M, VGPR0/1 = K0/K1;
//                                lanes 16-31 -> M, VGPR0/1 = K2/K3.
// B-slab layout (32-bit B 4x16): mirrored striping, N across lanes.
// C/D layout (32-bit 16x16): VGPR v -> M=v (lanes 0-15), M=v+8 (lanes 16-31).
// ---------------------------------------------------------------------------
__global__ __launch_bounds__(256) void finalize_wmma(
    float* __restrict__ out,            // sums in, averages out (in-place)
    const float* __restrict__ counts, int nseg) {
  const int lane = threadIdx.x & 31;
  const int wave = threadIdx.x >> 5;
  const int segtile = blockIdx.x * (blockDim.x >> 5) + wave;
  const int segbase = segtile * 16;
  if (segbase >= nseg) return;        // wave-uniform; EXEC stays all-ones

  const int m = lane & 15;            // M (segment within tile) / N (channel)
  const int half = lane >> 4;         // 0: K0/K1 pair, 1: K2/K3 pair

  float cnt = counts[segbase + m];
  float inv = 1.0f / fmaxf(cnt, 1.0f);

#pragma unroll
  for (int ctile = 0; ctile < 2; ++ctile) {
    v8f acc = {0.f, 0.f, 0.f, 0.f, 0.f, 0.f, 0.f, 0.f};
#pragma unroll
    for (int k = 0; k < 4; ++k) {
      const int kk0 = 4 * k + 2 * half;   // global K column of this lane's pair
      v2f a, b;
      // A = K-slab of diag(inv): A[m][kk] = (m == kk) ? inv[m] : 0
      a.x = (m == kk0) ? inv : 0.0f;
      a.y = (m == kk0 + 1) ? inv : 0.0f;
      // B = rows (segments) 4k..4k+3 of the sums tile, columns ctile*16+m
      b.x = out[(size_t)(segbase + kk0) * CHANNELS + ctile * 16 + m];
      b.y = out[(size_t)(segbase + kk0 + 1) * CHANNELS + ctile * 16 + m];
      acc = __builtin_amdgcn_wmma_f32_16x16x4_f32(
          /*neg_a=*/false, a, /*neg_b=*/false, b,
          /*c_mod=*/(short)0, acc, /*reuse_a=*/false, /*reuse_b=*/false);
    }
#pragma unroll
    for (int v = 0; v < 8; ++v) {
      out[(size_t)(segbase + v + 8 * half) * CHANNELS + ctile * 16 + m] =
          acc[v];
    }
  }
}

// ---------------------------------------------------------------------------
// Scatter phase: one row (pixel) per wave, lane == channel.
//   - 128B fully-coalesced row load per wave
//   - label broadcast via readfirstlane (scalar atomic base address)
//   - native no-return global_atomic_add_f32 (inline asm, SCOPE_DEV)
//   - global_prefetch_b8 to hide the grid-stride latency
// ---------------------------------------------------------------------------
__global__ __launch_bounds__(256) void scatter_add(
    const float* __restrict__ x, const int* __restrict__ sp,
    float* __restrict__ sums, float* __restrict__ counts, int nrows) {
  const int lane = threadIdx.x & 31;
  const int wid = (blockIdx.x * blockDim.x + threadIdx.x) >> 5;
  const int nw = (gridDim.x * blockDim.x) >> 5;

  // Per-wave streaming pointers; advance by the wave stride each iteration.
  const float* xp = x + (size_t)wid * CHANNELS + lane;
  const size_t xstep = (size_t)nw * CHANNELS;

  for (int row = wid; row < nrows; row += nw, xp += xstep) {
    if (row + nw < nrows) {
      __builtin_prefetch(xp + xstep, 0, 0);   // global_prefetch_b8
    }
    int seg = __builtin_amdgcn_readfirstlane(sp[row]);
    float v = *xp;                            // coalesced 128B/wave
    atomic_add_f32_dev(&sums[(size_t)seg * CHANNELS + lane], v);
    if (lane == 0) {
      atomic_add_f32_dev(&counts[seg], 1.0f);
    }
  }
}

// ---------------------------------------------------------------------------
// Host-side launcher. Inputs: [0] x (N*C f32), [1] superpixels (N int32),
// [2] num_segments (scalar, derived instead from out_size). Output: [S, C] f32.
// ---------------------------------------------------------------------------
extern "C" void kernel_launch(void* const* d_in, const int* in_sizes, int n_in,
                              void* d_out, int out_size, void* d_ws,
                              size_t ws_size, hipStream_t stream) {
  const float* x = (const float*)d_in[0];
  const int* sp = (const int*)d_in[1];
  float* out = (float*)d_out;            // accumulates sums, then averages
  float* counts = (float*)d_ws;          // [nseg] f32 counts
  const int nrows = in_sizes[1];         // N = 2,097,152
  const int nseg = out_size / CHANNELS;  // S = 8192

  // Zero accumulators (d_out poisoned by harness; must re-zero every call).
  zero_f32<<<256, 256, 0, stream>>>(out, out_size);
  zero_f32<<<32, 256, 0, stream>>>(counts, nseg);

  // Scatter-add: 2048 blocks x 256 threads = 16384 waves, ~128 rows each.
  scatter_add<<<2048, 256, 0, stream>>>(x, sp, out, counts, nrows);

  // Finalize: 16 segments per wave, 8 waves per block.
  const int ftiles = (nseg + 15) / 16;       // 512
  const int fblocks = (ftiles + 7) / 8;      // 64
  finalize_wmma<<<fblocks, 256, 0, stream>>>(out, counts, nseg);
}